// GLADMamba_22101901705746
// MI455X (gfx1250) — compile-verified
//
#include <hip/hip_runtime.h>
#include <hip/hip_bf16.h>
#include <math.h>

typedef float v2f __attribute__((ext_vector_type(2)));
typedef float v8f __attribute__((ext_vector_type(8)));

#define NNODES 4096
#define NEDGES 65536
#define NGRAPHS 512
#define DMODEL 512
#define EPSV 1e-5f

__device__ __forceinline__ float siluf(float x) { return x / (1.f + expf(-x)); }

__device__ __forceinline__ float wave_sum32(float v) {
  for (int off = 16; off > 0; off >>= 1) v += __shfl_xor(v, off, 32);
  return v;
}

// ---------------------------------------------------------------------------
// WMMA fp32 GEMM:  C[M,N] = act( (A (+A2)) @ W^T + bias )
// W stored row-major (N,K).  One wave computes a 16(M) x 64(N) strip using
// V_WMMA_F32_16X16X4_F32; block = 8 waves stacked over M (128 rows/block).
// Requires: M%128==0, N%64==0, K%4==0 (true for every GEMM in this net).
// ---------------------------------------------------------------------------
__global__ __launch_bounds__(256) void k_gemm(
    int M, int N, int K,
    const float* __restrict__ A, int lda,
    const float* __restrict__ A2, int lda2,
    const float* __restrict__ W,
    const float* __restrict__ bias,
    float* __restrict__ C, int ldc, int act)
{
  const int lane = threadIdx.x & 31;
  const int wid  = threadIdx.x >> 5;
  const int m0 = blockIdx.y * 128 + wid * 16;
  const int n0 = blockIdx.x * 64;
  if (m0 >= M) return;
  const int hi = lane >> 4;      // 0/1 -> K sub-pair
  const int lo = lane & 15;      // M (for A) / N (for B) index

  const v8f vz = {0.f,0.f,0.f,0.f,0.f,0.f,0.f,0.f};
  v8f acc[4] = {vz, vz, vz, vz};

  const int m = m0 + lo;
  const float* __restrict__ ar  = A + (size_t)m * lda;
  const float* __restrict__ ar2 = A2 ? (A2 + (size_t)m * lda2) : nullptr;
  const float* __restrict__ w0  = W + (size_t)(n0 + lo) * K;
  const size_t wstep = (size_t)16 * K;

  if (ar2) {
    for (int k = 0; k < K; k += 4) {
      const int kb = k + 2 * hi;
      v2f a; a.x = ar[kb] + ar2[kb]; a.y = ar[kb + 1] + ar2[kb + 1];
      v2f b0; b0.x = w0[kb];             b0.y = w0[kb + 1];
      v2f b1; b1.x = w0[wstep + kb];     b1.y = w0[wstep + kb + 1];
      v2f b2; b2.x = w0[2*wstep + kb];   b2.y = w0[2*wstep + kb + 1];
      v2f b3; b3.x = w0[3*wstep + kb];   b3.y = w0[3*wstep + kb + 1];
      acc[0] = __builtin_amdgcn_wmma_f32_16x16x4_f32(false, a, false, b0, (short)0, acc[0], false, false);
      acc[1] = __builtin_amdgcn_wmma_f32_16x16x4_f32(false, a, false, b1, (short)0, acc[1], false, false);
      acc[2] = __builtin_amdgcn_wmma_f32_16x16x4_f32(false, a, false, b2, (short)0, acc[2], false, false);
      acc[3] = __builtin_amdgcn_wmma_f32_16x16x4_f32(false, a, false, b3, (short)0, acc[3], false, false);
    }
  } else {
    for (int k = 0; k < K; k += 4) {
      const int kb = k + 2 * hi;
      v2f a; a.x = ar[kb]; a.y = ar[kb + 1];
      v2f b0; b0.x = w0[kb];             b0.y = w0[kb + 1];
      v2f b1; b1.x = w0[wstep + kb];     b1.y = w0[wstep + kb + 1];
      v2f b2; b2.x = w0[2*wstep + kb];   b2.y = w0[2*wstep + kb + 1];
      v2f b3; b3.x = w0[3*wstep + kb];   b3.y = w0[3*wstep + kb + 1];
      acc[0] = __builtin_amdgcn_wmma_f32_16x16x4_f32(false, a, false, b0, (short)0, acc[0], false, false);
      acc[1] = __builtin_amdgcn_wmma_f32_16x16x4_f32(false, a, false, b1, (short)0, acc[1], false, false);
      acc[2] = __builtin_amdgcn_wmma_f32_16x16x4_f32(false, a, false, b2, (short)0, acc[2], false, false);
      acc[3] = __builtin_amdgcn_wmma_f32_16x16x4_f32(false, a, false, b3, (short)0, acc[3], false, false);
    }
  }

  // C/D layout: VGPR r -> row m0+r (lanes 0-15) / m0+8+r (lanes 16-31), col = n-tile + lane%16
  const int rowBase = m0 + 8 * hi;
  for (int j = 0; j < 4; ++j) {
    const int col = n0 + 16 * j + lo;
    const float bv = bias ? bias[col] : 0.f;
    for (int r = 0; r < 8; ++r) {
      float v = acc[j][r] + bv;
      if      (act == 1) v = fmaxf(v, 0.f);
      else if (act == 2) v = (v > 0.f) ? v : 0.01f * v;
      else if (act == 3) v = (v > 20.f) ? v : log1pf(expf(v));   // softplus
      C[(size_t)(rowBase + r) * ldc + col] = v;
    }
  }
}

// --------------------------- GIN support -----------------------------------
__global__ __launch_bounds__(256) void k_scatter_edges(
    const float* __restrict__ x, int ldx,
    const int* __restrict__ src, const int* __restrict__ tgt,
    float* __restrict__ agg, int din)
{
  int t = blockIdx.x * blockDim.x + threadIdx.x;
  if (t >= NEDGES * din) return;
  int e = t / din, c = t - e * din;
  atomicAdd(&agg[(size_t)tgt[e] * din + c], x[(size_t)src[e] * ldx + c]);
}

__global__ __launch_bounds__(256) void k_count(const int* __restrict__ batch, float* __restrict__ cnt)
{
  int i = blockIdx.x * blockDim.x + threadIdx.x;
  if (i < NNODES) atomicAdd(&cnt[batch[i]], 1.f);
}

__global__ __launch_bounds__(256) void k_pool(
    const float* __restrict__ ncat, const int* __restrict__ batch, float* __restrict__ gcat)
{
  int t = blockIdx.x * blockDim.x + threadIdx.x;
  if (t >= NNODES * DMODEL) return;
  int v = t >> 9, c = t & 511;
  atomicAdd(&gcat[(size_t)batch[v] * DMODEL + c], ncat[t]);
}

__global__ __launch_bounds__(256) void k_pool_scale(float* __restrict__ gcat, const float* __restrict__ cnt)
{
  int t = blockIdx.x * blockDim.x + threadIdx.x;
  if (t >= NGRAPHS * DMODEL) return;
  gcat[t] /= fmaxf(cnt[t >> 9], 1.f);
}

// --------------------------- norms -----------------------------------------
__global__ __launch_bounds__(256) void k_rmsnorm(
    const float* __restrict__ x, const float* __restrict__ w, float* __restrict__ y, int rows)
{
  int row = blockIdx.x * (blockDim.x >> 5) + (threadIdx.x >> 5);
  if (row >= rows) return;
  int lane = threadIdx.x & 31;
  const float* xr = x + (size_t)row * DMODEL;
  float v[16]; float s = 0.f;
  for (int i = 0; i < 16; ++i) { v[i] = xr[lane + 32 * i]; s += v[i] * v[i]; }
  s = wave_sum32(s);
  float r = rsqrtf(s * (1.f / DMODEL) + EPSV);
  float* yr = y + (size_t)row * DMODEL;
  for (int i = 0; i < 16; ++i) { int c = lane + 32 * i; yr[c] = v[i] * r * w[c]; }
}

// out = rmsnorm( x1 + rmsnorm(outg, nw), onw )   (fused mamba epilogue)
__global__ __launch_bounds__(256) void k_finalize(
    const float* __restrict__ outg, const float* __restrict__ x1,
    const float* __restrict__ nw, const float* __restrict__ onw,
    float* __restrict__ dst, int rows)
{
  int row = blockIdx.x * (blockDim.x >> 5) + (threadIdx.x >> 5);
  if (row >= rows) return;
  int lane = threadIdx.x & 31;
  const float* o = outg + (size_t)row * DMODEL;
  const float* x = x1   + (size_t)row * DMODEL;
  float ov[16], zv[16]; float s1 = 0.f;
  for (int i = 0; i < 16; ++i) { float v = o[lane + 32 * i]; ov[i] = v; s1 += v * v; }
  s1 = wave_sum32(s1);
  float r1 = rsqrtf(s1 * (1.f / DMODEL) + EPSV);
  float s2 = 0.f;
  for (int i = 0; i < 16; ++i) {
    int c = lane + 32 * i;
    float z = x[c] + ov[i] * r1 * nw[c];
    zv[i] = z; s2 += z * z;
  }
  s2 = wave_sum32(s2);
  float r2 = rsqrtf(s2 * (1.f / DMODEL) + EPSV);
  float* d = dst + (size_t)row * DMODEL;
  for (int i = 0; i < 16; ++i) { int c = lane + 32 * i; d[c] = zv[i] * r2 * onw[c]; }
}

// --------------------------- mamba helpers ---------------------------------
// causal depthwise conv on an L-broadcast input collapses to 4 per-channel
// scalars: S[c][t] = sum of conv taps visible at position t (t>=3 identical).
__global__ __launch_bounds__(256) void k_prep(
    const float* __restrict__ conv_w, const float* __restrict__ A_log,
    float* __restrict__ Sc, float* __restrict__ An)
{
  int d = blockIdx.x * blockDim.x + threadIdx.x;
  if (d >= DMODEL) return;
  float w0 = conv_w[d * 4 + 0], w1 = conv_w[d * 4 + 1], w2 = conv_w[d * 4 + 2], w3 = conv_w[d * 4 + 3];
  Sc[d * 4 + 0] = w3;
  Sc[d * 4 + 1] = w2 + w3;
  Sc[d * 4 + 2] = w1 + w2 + w3;
  Sc[d * 4 + 3] = w0 + w1 + w2 + w3;
  for (int n = 0; n < 16; ++n) An[d * 16 + n] = -expf(A_log[d * 16 + n]);
}

// xc2s[lp,b,d] = silu(xc2p[b,d]*S[d][lp] + conv_b[d])  for lp=0..3
__global__ __launch_bounds__(256) void k_xc2s(
    const float* __restrict__ xc2p, const float* __restrict__ Sc,
    const float* __restrict__ cb, float* __restrict__ xc2s, int b)
{
  int t = blockIdx.x * blockDim.x + threadIdx.x;
  if (t >= b * DMODEL) return;
  int bi = t >> 9, d = t & 511;
  float x = xc2p[t], cbv = cb[d];
  for (int lp = 0; lp < 4; ++lp)
    xc2s[((size_t)lp * b + bi) * DMODEL + d] = siluf(x * Sc[d * 4 + lp] + cbv);
}

// selective scan over L_SEQ=6, D_STATE=16, only last-position output needed.
__global__ __launch_bounds__(256) void k_scan(
    const float* __restrict__ xr1,    // b x 1024 (xc1 | res1)
    const float* __restrict__ delta,  // 4b x 512 (lp-major)
    const float* __restrict__ dbl,    // 4b x 64  (delta_in | B | C)
    const float* __restrict__ Sc, const float* __restrict__ cb,
    const float* __restrict__ An,     // 512 x 16, negated A
    const float* __restrict__ Dp,
    float* __restrict__ ymam, int b)
{
  int t = blockIdx.x * blockDim.x + threadIdx.x;
  if (t >= b * DMODEL) return;
  int bi = t >> 9, d = t & 511;
  float xc1 = xr1[(size_t)bi * 1024 + d];
  float res = xr1[(size_t)bi * 1024 + DMODEL + d];
  float s[4];
  for (int lp = 0; lp < 4; ++lp) s[lp] = Sc[d * 4 + lp];
  float cbv = cb[d];
  float Av[16];
  for (int n = 0; n < 16; ++n) Av[n] = An[d * 16 + n];
  float h[16];
  for (int n = 0; n < 16; ++n) h[n] = 0.f;
  float u = 0.f;
  for (int l = 0; l < 6; ++l) {
    int lp = l < 3 ? l : 3;
    float dl = delta[((size_t)lp * b + bi) * DMODEL + d];
    u = siluf(xc1 * s[lp] + cbv);
    float du = dl * u;
    const float* Bp = dbl + ((size_t)lp * b + bi) * 64 + 32;
    for (int n = 0; n < 16; ++n) h[n] = expf(dl * Av[n]) * h[n] + du * Bp[n];
  }
  const float* Cp = dbl + ((size_t)3 * b + bi) * 64 + 48;
  float y = 0.f;
  for (int n = 0; n < 16; ++n) y += h[n] * Cp[n];
  y += u * Dp[d];          // u holds position-5 value
  y *= siluf(res);
  ymam[t] = y;
}

// ===========================================================================
// Host side
// ===========================================================================
struct PP {
  const float *x[2], *XLX[2];
  const float *gw1[2][4], *gb1[2][4], *gw2[2][4], *gb2[2][4];
  const float *hw1[4], *hb1[4], *hw2[4], *hb2[4];   // pg_f, pg_s, pn_f, pn_s
  const float *rqw[2], *rqb[2];
  const float *norm_w, *conv_w, *conv_b, *in_w, *x_w, *dt_w, *dt_b, *out_w, *A_log, *Dvec, *out_norm_w;
  const int *ei, *batch;
};

static inline void gemm(hipStream_t s, int M, int N, int K,
                        const float* A, int lda, const float* A2, int lda2,
                        const float* W, const float* bias, float* C, int ldc, int act)
{
  dim3 g(N / 64, M / 128);
  k_gemm<<<g, dim3(256), 0, s>>>(M, N, K, A, lda, A2, lda2, W, bias, C, ldc, act);
}

extern "C" void kernel_launch(void* const* d_in, const int* in_sizes, int n_in,
                              void* d_out, int out_size, void* d_ws, size_t ws_size,
                              hipStream_t stream)
{
  (void)n_in; (void)out_size; (void)ws_size;
  auto F = [&](int i) { return (const float*)d_in[i]; };
  auto I = [&](int i) { return (const int*)d_in[i]; };

  PP p;
  if (in_sizes[0] == NNODES * 64) {
    // insertion-order flattening
    p.x[0] = F(0); p.x[1] = F(1); p.XLX[0] = F(2); p.XLX[1] = F(3);
    int k = 4;
    for (int e = 0; e < 2; ++e)
      for (int l = 0; l < 4; ++l) {
        p.gw1[e][l] = F(k); p.gb1[e][l] = F(k + 1); p.gw2[e][l] = F(k + 2); p.gb2[e][l] = F(k + 3);
        k += 4;
      }
    for (int h = 0; h < 4; ++h) {
      p.hw1[h] = F(k); p.hb1[h] = F(k + 1); p.hw2[h] = F(k + 2); p.hb2[h] = F(k + 3);
      k += 4;
    }
    p.rqw[0] = F(52); p.rqb[0] = F(53); p.rqw[1] = F(54); p.rqb[1] = F(55);
    p.norm_w = F(56); p.conv_w = F(57); p.conv_b = F(58); p.in_w = F(59); p.x_w = F(60);
    p.dt_w = F(61); p.dt_b = F(62); p.out_w = F(63); p.A_log = F(64); p.Dvec = F(65);
    p.out_norm_w = F(66);
    p.ei = I(67); p.batch = I(68);
  } else {
    // sorted (jax tree_flatten) flattening
    p.XLX[0] = F(0); p.XLX[1] = F(1); p.batch = I(2); p.ei = I(3);
    int k = 5;
    for (int e = 0; e < 2; ++e)
      for (int l = 0; l < 4; ++l) {
        p.gb1[e][l] = F(k); p.gb2[e][l] = F(k + 1); p.gw1[e][l] = F(k + 2); p.gw2[e][l] = F(k + 3);
        k += 4;
      }
    p.A_log = F(37); p.Dvec = F(38); p.conv_b = F(39); p.conv_w = F(40); p.dt_b = F(41);
    p.dt_w = F(42); p.in_w = F(43); p.norm_w = F(44); p.out_w = F(45); p.x_w = F(46);
    p.out_norm_w = F(47);
    k = 48;
    for (int h = 0; h < 4; ++h) {
      p.hb1[h] = F(k); p.hb2[h] = F(k + 1); p.hw1[h] = F(k + 2); p.hw2[h] = F(k + 3);
      k += 4;
    }
    p.rqb[0] = F(64); p.rqw[0] = F(65); p.rqb[1] = F(66); p.rqw[1] = F(67);
    p.x[0] = F(68); p.x[1] = F(69);
  }

  // ---------------- workspace layout (floats) ------------------------------
  float* ws = (float*)d_ws;
  size_t off = 0;
  auto take = [&](size_t n) { float* r = ws + off; off += n; return r; };
  const size_t NM = (size_t)NNODES * DMODEL;      // 2097152
  const size_t GM = (size_t)NGRAPHS * DMODEL;     // 262144
  float* n1[2]; n1[0] = take(NM); n1[1] = take(NM);
  float* g1[2]; g1[0] = take(GM); g1[1] = take(GM);
  float* RQ[2]; RQ[0] = take(GM); RQ[1] = take(GM);
  float* Sc = take(DMODEL * 4);
  float* An = take(DMODEL * 16);
  float* cnt = take(NGRAPHS);
  size_t arena = off;
  // GIN-phase view of arena
  float* ncat[2]; ncat[0] = ws + arena; ncat[1] = ncat[0] + NM;
  float* gcat[2]; gcat[0] = ncat[1] + NM; gcat[1] = gcat[0] + GM;
  float* agg   = gcat[1] + GM;                    // 4096*128
  float* h1    = agg + (size_t)NNODES * 128;      // 4096*128
  float* headt = h1 + (size_t)NNODES * 128;       // 4096*512
  // mamba-phase view of arena (reuses GIN buffers; heads finished by then)
  float* x1n  = ws + arena;
  float* x2n  = x1n + NM;
  float* xr1  = x2n + NM;                          // 4096*1024
  float* xc2p = xr1 + (size_t)NNODES * 1024;
  float* xc2s = xc2p + NM;                         // 4*4096*512
  float* dbl  = xc2s + 4 * NM;                     // 4*4096*64
  float* dly  = dbl + (size_t)4 * NNODES * 64;     // 4*4096*512
  float* ymam = dly + 4 * NM;
  float* outg = ymam + NM;

  const int* src = p.ei;
  const int* tgt = p.ei + NEDGES;

  // ---------------- global precompute --------------------------------------
  k_prep<<<2, 256, 0, stream>>>(p.conv_w, p.A_log, Sc, An);
  hipMemsetAsync(cnt, 0, NGRAPHS * sizeof(float), stream);
  k_count<<<(NNODES + 255) / 256, 256, 0, stream>>>(p.batch, cnt);

  // ---------------- GIN encoders -------------------------------------------
  for (int e = 0; e < 2; ++e) {
    for (int l = 0; l < 4; ++l) {
      const float* xin = (l == 0) ? p.x[e] : (ncat[e] + 128 * (l - 1));
      int ldx = (l == 0) ? 64 : DMODEL;
      int din = (l == 0) ? 64 : 128;
      hipMemsetAsync(agg, 0, (size_t)NNODES * din * sizeof(float), stream);
      int nt = NEDGES * din;
      k_scatter_edges<<<(nt + 255) / 256, 256, 0, stream>>>(xin, ldx, src, tgt, agg, din);
      // h1 = relu((x+agg) @ w1^T + b1)
      gemm(stream, NNODES, 128, din, xin, ldx, agg, din, p.gw1[e][l], p.gb1[e][l], h1, 128, 1);
      // x' = relu(h1 @ w2^T + b2) -> concat slot
      gemm(stream, NNODES, 128, 128, h1, 128, nullptr, 0, p.gw2[e][l], p.gb2[e][l], ncat[e] + 128 * l, DMODEL, 1);
    }
    hipMemsetAsync(gcat[e], 0, GM * sizeof(float), stream);
    k_pool<<<(int)((NM + 255) / 256), 256, 0, stream>>>(ncat[e], p.batch, gcat[e]);
    k_pool_scale<<<(int)((GM + 255) / 256), 256, 0, stream>>>(gcat[e], cnt);
  }

  // ---------------- MLP heads ----------------------------------------------
  const float* hin[4] = {gcat[0], gcat[1], ncat[0], ncat[1]};
  float* hout[4]      = {g1[0],   g1[1],   n1[0],   n1[1]};
  int   hM[4]         = {NGRAPHS, NGRAPHS, NNODES,  NNODES};
  for (int h = 0; h < 4; ++h) {
    gemm(stream, hM[h], DMODEL, DMODEL, hin[h], DMODEL, nullptr, 0, p.hw1[h], p.hb1[h], headt, DMODEL, 1);
    gemm(stream, hM[h], DMODEL, DMODEL, headt,  DMODEL, nullptr, 0, p.hw2[h], p.hb2[h], hout[h], DMODEL, 0);
  }
  // RQ = leaky_relu(XLX @ rq_w^T + rq_b, 0.01)
  for (int e = 0; e < 2; ++e)
    gemm(stream, NGRAPHS, DMODEL, 64, p.XLX[e], 64, nullptr, 0, p.rqw[e], p.rqb[e], RQ[e], DMODEL, 2);

  // ---------------- 4 mamba blocks -----------------------------------------
  float* out = (float*)d_out;
  const float* mx1[4] = {g1[0], g1[1], n1[0], n1[1]};
  const float* mx2[4] = {RQ[0], RQ[1], n1[1], n1[0]};
  int   mb[4]  = {NGRAPHS, NGRAPHS, NNODES, NNODES};
  size_t mo[4] = {0, GM, 2 * GM, 2 * GM + NM};

  for (int c = 0; c < 4; ++c) {
    int b = mb[c];
    int rows8 = (b + 7) / 8;
    int ev = (b * DMODEL + 255) / 256;
    k_rmsnorm<<<rows8, 256, 0, stream>>>(mx1[c], p.norm_w, x1n, b);
    k_rmsnorm<<<rows8, 256, 0, stream>>>(mx2[c], p.norm_w, x2n, b);
    // xr1 = x1n @ in_w^T  (b x 1024); xc2p = x2n @ in_w[:512]^T
    gemm(stream, b, 2 * DMODEL, DMODEL, x1n, DMODEL, nullptr, 0, p.in_w, nullptr, xr1, 2 * DMODEL, 0);
    gemm(stream, b, DMODEL,     DMODEL, x2n, DMODEL, nullptr, 0, p.in_w, nullptr, xc2p, DMODEL, 0);
    k_xc2s<<<ev, 256, 0, stream>>>(xc2p, Sc, p.conv_b, xc2s, b);
    // dbl = xc2s @ x_w^T   (4b x 64)
    gemm(stream, 4 * b, 64, DMODEL, xc2s, DMODEL, nullptr, 0, p.x_w, nullptr, dbl, 64, 0);
    // delta = softplus(dbl[:, :32] @ dt_w^T + dt_b)  (4b x 512)
    gemm(stream, 4 * b, DMODEL, 32, dbl, 64, nullptr, 0, p.dt_w, p.dt_b, dly, DMODEL, 3);
    k_scan<<<ev, 256, 0, stream>>>(xr1, dly, dbl, Sc, p.conv_b, An, p.Dvec, ymam, b);
    // out = ymam @ out_w^T, then fused rmsnorm + residual + final rmsnorm
    gemm(stream, b, DMODEL, DMODEL, ymam, DMODEL, nullptr, 0, p.out_w, nullptr, outg, DMODEL, 0);
    k_finalize<<<rows8, 256, 0, stream>>>(outg, mx1[c], p.norm_w, p.out_norm_w, out + mo[c], b);
  }
}